// Loss_71382356460152
// MI455X (gfx1250) — compile-verified
//
#include <hip/hip_runtime.h>
#include <math.h>

typedef __attribute__((ext_vector_type(2))) float v2f;
typedef __attribute__((ext_vector_type(8))) float v8f;

#define NUM_CLASSES 80
#define EPS_F 1e-7f

__device__ __forceinline__ float sigmoidf(float x) {
    return 1.0f / (1.0f + expf(-x));
}

// bce with target==1 : -log(clip(p))
__device__ __forceinline__ float bce1(float p) {
    p = fminf(fmaxf(p, EPS_F), 1.0f - EPS_F);
    return -logf(p);
}
// bce with target==0 : -log(1 - clip(p))
__device__ __forceinline__ float bce0(float p) {
    p = fminf(fmaxf(p, EPS_F), 1.0f - EPS_F);
    return -logf(1.0f - p);
}

// Wave32 all-lane sum using the f32 WMMA unit:
//   A(16x4): lanes 0-15 row m, VGPR0=K0; lanes 16-31 row m, VGPR0=K2. a.y=0.
//   B(4x16) = ones  =>  D[m][n] = v_m + v_{m+16} replicated over n.
//   C/D layout: c[r] on lane<16 = row r, on lane>=16 = row r+8.
//   sum(c[0..7]) + shfl_xor(16) = full 32-lane sum, on every lane.
__device__ __forceinline__ float wave_sum32(float v) {
#if defined(__has_builtin) && __has_builtin(__builtin_amdgcn_wmma_f32_16x16x4_f32)
    v2f a; a.x = v;    a.y = 0.0f;
    v2f b; b.x = 1.0f; b.y = 1.0f;
    v8f c = {};
    c = __builtin_amdgcn_wmma_f32_16x16x4_f32(false, a, false, b,
                                              (short)0, c, false, false);
    float s = c[0] + c[1] + c[2] + c[3] + c[4] + c[5] + c[6] + c[7];
    s += __shfl_xor(s, 16, 32);
    return s;
#else
    for (int off = 16; off > 0; off >>= 1) v += __shfl_xor(v, off, 32);
    return v;
#endif
}

__global__ void zero_kernel(unsigned* __restrict__ p, int n) {
    int i = blockIdx.x * blockDim.x + threadIdx.x;
    if (i < n) p[i] = 0u;
}

// One thread per target: anchor-IoU argmax, ignore-bit scatter, cell ownership.
__global__ void decode_kernel(const float* __restrict__ targets,
                              unsigned* __restrict__ owner,
                              unsigned* __restrict__ ignbits,
                              float* __restrict__ tinfo,
                              int B, int T, int H, int W,
                              float aw0, float ah0, float aw1, float ah1,
                              float aw2, float ah2) {
    int tid = blockIdx.x * blockDim.x + threadIdx.x;
    if (tid >= B * T) return;
    const float* tg = targets + (size_t)tid * 5;
    float c = tg[0], x = tg[1], y = tg[2], w = tg[3], h = tg[4];
    if (c + x + y + w + h == 0.0f) return;   // invalid target -> OOB scatter dropped

    float gx = x * (float)W, gy = y * (float)H;
    float gw = w * (float)W, gh = h * (float)H;
    int gi = (int)gx; gi = min(max(gi, 0), W - 1);
    int gj = (int)gy; gj = min(max(gj, 0), H - 1);
    int cls = (int)c;
    int b = tid / T;

    float AW[3] = {aw0, aw1, aw2};
    float AH[3] = {ah0, ah1, ah2};
    float area = gw * gh;
    int best = 0;
    float bestIou = -1.0f;
#pragma unroll
    for (int a = 0; a < 3; ++a) {
        float inter = fminf(gw, AW[a]) * fminf(gh, AH[a]);
        float iou = inter / (area + AW[a] * AH[a] - inter + 1e-16f);
        if (iou > bestIou) { bestIou = iou; best = a; }   // first max wins (argmax)
        if (iou > 0.5f) {                                  // noobj <- 0 at (b,a,gj,gi)
            int cell = ((b * 3 + a) * H + gj) * W + gi;
            atomicOr(&ignbits[cell >> 5], 1u << (cell & 31));
        }
    }
    int cell = ((b * 3 + best) * H + gj) * W + gi;
    atomicMax(&owner[cell], (unsigned)(tid + 1));   // deterministic dedupe
    float* ti = tinfo + (size_t)tid * 8;
    ti[0] = gx; ti[1] = gy; ti[2] = gw; ti[3] = gh; ti[4] = (float)cls;
}

// One thread per (b, anchor, j, i) cell. Dense conf term; sparse iou/cls terms
// only at owned cells. WMMA wave reduction + LDS block reduction + atomics.
__global__ void dense_kernel(const float* __restrict__ input,
                             const unsigned* __restrict__ owner,
                             const unsigned* __restrict__ ignbits,
                             const float* __restrict__ tinfo,
                             float* __restrict__ acc,
                             int B, int H, int W,
                             float aw0, float ah0, float aw1, float ah1,
                             float aw2, float ah2) {
    const int HW = H * W;
    const int NCELL = B * 3 * HW;
    int idx = blockIdx.x * blockDim.x + threadIdx.x;

    float confPart = 0.0f, iouPart = 0.0f, clsPart = 0.0f, nposPart = 0.0f;
    if (idx < NCELL) {
        int i = idx % W;
        int j = (idx / W) % H;
        int a = (idx / HW) % 3;
        int b = idx / (3 * HW);
        size_t base = (size_t)b * 255 * HW + (size_t)a * 85 * HW
                    + (size_t)j * W + (size_t)i;

        float conf = sigmoidf(input[base + 4 * (size_t)HW]);
        unsigned own = owner[idx];
        bool ign = (ignbits[idx >> 5] >> (idx & 31)) & 1u;

        const float C0 = -logf(1.0f - EPS_F);   // bce(clip(0), 0)
        float t1 = own ? bce1(conf) : C0;       // bce(conf*mask, mask)
        float t2 = ign ? C0 : bce0(conf);       // bce(conf*noobj, 0)
        confPart = t1 + 0.5f * t2;

        if (own) {
            const float* ti = tinfo + (size_t)(own - 1) * 8;
            float kx = ti[0], ky = ti[1], kw = ti[2], kh = ti[3];
            int cls = (int)ti[4];
            float AW[3] = {aw0, aw1, aw2};
            float AH[3] = {ah0, ah1, ah2};

            float px = sigmoidf(input[base]);
            float py = sigmoidf(input[base + (size_t)HW]);
            float pw = expf(input[base + 2 * (size_t)HW]) * AW[a];
            float ph = expf(input[base + 3 * (size_t)HW]) * AH[a];
            float hx = px + (float)i;
            float hy = py + (float)j;

            float iw = fmaxf(fminf(hx + pw * 0.5f, kx + kw * 0.5f) -
                             fmaxf(hx - pw * 0.5f, kx - kw * 0.5f), 0.0f);
            float ih = fmaxf(fminf(hy + ph * 0.5f, ky + kh * 0.5f) -
                             fmaxf(hy - ph * 0.5f, ky - kh * 0.5f), 0.0f);
            float inter = iw * ih;
            float iou = inter / (pw * ph + kw * kh - inter + 1e-16f);
            iouPart = 1.0f - iou;
            nposPart = 1.0f;

            float s = 0.0f;
            for (int cc = 0; cc < NUM_CLASSES; ++cc) {
                float p = sigmoidf(input[base + (size_t)(5 + cc) * HW]);
                s += (cc == cls) ? bce1(p) : bce0(p);
            }
            clsPart = s;
        }
    }

    // All 256 threads reach here (no early return) -> EXEC all ones for WMMA.
    float s0 = wave_sum32(confPart);
    float s1 = wave_sum32(iouPart);
    float s2 = wave_sum32(clsPart);
    float s3 = wave_sum32(nposPart);

    __shared__ float red[8][4];
    int wave = threadIdx.x >> 5;
    int lane = threadIdx.x & 31;
    if (lane == 0) {
        red[wave][0] = s0; red[wave][1] = s1;
        red[wave][2] = s2; red[wave][3] = s3;
    }
    __syncthreads();
    if (threadIdx.x < 4) {
        float t = 0.0f;
        int nw = (int)(blockDim.x >> 5);
        for (int wv = 0; wv < nw; ++wv) t += red[wv][threadIdx.x];
        atomicAdd(&acc[threadIdx.x], t);
    }
}

__global__ void finalize_kernel(const float* __restrict__ acc,
                                float* __restrict__ out, int NCELL) {
    if (threadIdx.x == 0 && blockIdx.x == 0) {
        float sumConf = acc[0], sumIou = acc[1], sumCls = acc[2], npos = acc[3];
        float lossConf = sumConf / (float)NCELL;
        float lossIou  = sumIou;
        float lossCls  = sumCls / (fmaxf(npos, 1.0f) * (float)NUM_CLASSES);
        float loss = 0.5f * lossIou + lossConf + lossCls;
        out[0] = loss; out[1] = lossIou; out[2] = lossConf; out[3] = lossCls;
    }
}

extern "C" void kernel_launch(void* const* d_in, const int* in_sizes, int n_in,
                              void* d_out, int out_size, void* d_ws, size_t ws_size,
                              hipStream_t stream) {
    const float* input   = (const float*)d_in[0];   // (B, 255, H, W) f32
    const float* targets = (const float*)d_in[1];   // (B, 50, 5)     f32

    const int T = 50;
    int B = in_sizes[1] / (T * 5);
    long hw = (long)in_sizes[0] / ((long)B * 255);
    int W = (int)lround(sqrt((double)hw));          // square feature map (52)
    int H = W;
    float strideW = 416.0f / (float)W;
    float strideH = 416.0f / (float)H;
    float aw0 = 10.0f / strideW, ah0 = 13.0f / strideH;
    float aw1 = 16.0f / strideW, ah1 = 30.0f / strideH;
    float aw2 = 33.0f / strideW, ah2 = 23.0f / strideH;

    int HW = H * W;
    int NCELL = B * 3 * HW;
    int IGNW  = (NCELL + 31) / 32;

    // workspace layout (u32 words): [owner NCELL][ignbits IGNW][acc 8][tinfo B*T*8]
    unsigned* owner   = (unsigned*)d_ws;
    unsigned* ignbits = owner + NCELL;
    float*    acc     = (float*)(ignbits + IGNW);
    float*    tinfo   = acc + 8;

    int zeroWords = NCELL + IGNW + 8;
    zero_kernel<<<(zeroWords + 255) / 256, 256, 0, stream>>>(owner, zeroWords);
    decode_kernel<<<(B * T + 255) / 256, 256, 0, stream>>>(
        targets, owner, ignbits, tinfo, B, T, H, W,
        aw0, ah0, aw1, ah1, aw2, ah2);
    dense_kernel<<<(NCELL + 255) / 256, 256, 0, stream>>>(
        input, owner, ignbits, tinfo, acc, B, H, W,
        aw0, ah0, aw1, ah1, aw2, ah2);
    finalize_kernel<<<1, 32, 0, stream>>>(acc, (float*)d_out, NCELL);
}